// FNOBlocks_38010460570535
// MI455X (gfx1250) — compile-verified
//
#include <hip/hip_runtime.h>
#include <hip/hip_bf16.h>

// ---------------------------------------------------------------------------
// FNO block on gfx1250 (CDNA5): all DFT stages + channel MLP as f32 WMMA GEMMs
// (V_WMMA_F32_16X16X4_F32), spectral per-mode channel mix as coalesced VALU.
// ---------------------------------------------------------------------------

typedef float v2f __attribute__((ext_vector_type(2)));
typedef float v8f __attribute__((ext_vector_type(8)));

static __device__ __forceinline__ v8f v8zero() {
  v8f z = {0.f, 0.f, 0.f, 0.f, 0.f, 0.f, 0.f, 0.f};
  return z;
}

static __device__ __forceinline__ v8f wmma4(v2f a, v2f b, v8f c) {
  // D = A(16x4) * B(4x16) + C(16x16), fp32 matrix pipe
  return __builtin_amdgcn_wmma_f32_16x16x4_f32(false, a, false, b, (short)0, c,
                                               false, false);
}

// A-fragment (16x4 f32): lanes 0-15 hold row M=lane, K = k0+0..1 in v0,v1;
// lanes 16-31 hold row M=lane-16, K = k0+2..3.   (ISA 7.12.2)
static __device__ __forceinline__ v2f ldA(const float* __restrict__ A, int lda,
                                          int m0, int k0, int lane) {
  int half = (lane >> 4) & 1;
  int r = lane & 15;
  const float* p = A + (size_t)(m0 + r) * lda + k0 + 2 * half;
  v2f v;
  v.x = p[0];
  v.y = p[1];
  return v;
}

// B-fragment (4x16 f32): lane n in [0,15]: v0=B[k0][n], v1=B[k0+1][n];
// lanes 16-31: v0=B[k0+2][n-16], v1=B[k0+3][n-16]. (transpose-symmetric to A)
static __device__ __forceinline__ v2f ldB(const float* __restrict__ B, int ldb,
                                          int k0, int n0, int lane) {
  int half = (lane >> 4) & 1;
  int n = lane & 15;
  const float* p = B + (size_t)(k0 + 2 * half) * ldb + n0 + n;
  v2f v;
  v.x = p[0];
  v.y = p[ldb];
  return v;
}

// C/D fragment store: VGPR g -> row m0+g (lanes 0-15) / m0+g+8 (lanes 16-31),
// col n0 + (lane&15).
static __device__ __forceinline__ void stC(float* __restrict__ dst, int ldd,
                                           int m0, int n0, int lane, v8f c) {
  int half = (lane >> 4) & 1;
  int n = lane & 15;
#pragma unroll
  for (int g = 0; g < 8; ++g)
    dst[(size_t)(m0 + g + half * 8) * ldd + n0 + n] = c[g];
}

static __device__ __forceinline__ float gelu_exact(float v) {
  return 0.5f * v * (1.0f + erff(v * 0.70710678118654752f));
}

// ---------------------------------------------------------------------------
// Workspace layout (float offsets)
// ---------------------------------------------------------------------------
#define OFF_TAB 0u
// tables: Ffwd[128][64] @0, CH[64][128] @8192, SH @16384, CI[128][64] @24576,
//         SI @32768, FI[64][128] @40960
#define OFF_YFR 49152u                       // [B,C,H,32] re   (4,194,304)
#define OFF_YFI (OFF_YFR + 4194304u)         // [B,C,H,32] im
#define OFF_ZFR (OFF_YFI + 4194304u)         // [B,C,64,32] re  (2,097,152)
#define OFF_ZFI (OFF_ZFR + 2097152u)
#define OFF_MXR (OFF_ZFI + 2097152u)         // [B,O,64,32] re
#define OFF_MXI (OFF_MXR + 2097152u)
#define OFF_Y   (OFF_MXI + 2097152u)         // [B,C,H,W] spatial (16,777,216)
// inverse-H output [B,O,H,32] re/im aliases OFF_YFR / OFF_YFI (yf is dead)

// ---------------------------------------------------------------------------
// Kernel 0: twiddle tables (cheap, deterministic, graph-safe)
// ---------------------------------------------------------------------------
__global__ __launch_bounds__(256) void k_tables(float* __restrict__ ws) {
  const float PI2 = 6.28318530717958647692f;
  int tid = threadIdx.x;
  float* Ffwd = ws;               // [k=w 128][n 64] : n<32 cos, n>=32 -sin
  float* CH = ws + 8192;          // [j 64][h 128] cos
  float* SH = ws + 16384;         // [j 64][h 128] sin
  float* CI = ws + 24576;         // [h 128][j 64] cos/128
  float* SI = ws + 32768;         // [h 128][j 64] sin/128
  float* FI = ws + 40960;         // [k 64][w 128] irfft (1/W, doubling folded)
  for (int i = tid; i < 8192; i += 256) {
    int w = i >> 6, n = i & 63;
    float th = PI2 * (float)w * (float)(n & 31) * (1.0f / 128.0f);
    Ffwd[i] = (n < 32) ? __cosf(th) : -__sinf(th);
  }
  for (int i = tid; i < 8192; i += 256) {
    int j = i >> 7, h = i & 127;
    float f = (j < 32) ? (float)j : (float)(j + 64);  // kh in {0..31,96..127}
    float th = PI2 * (float)h * f * (1.0f / 128.0f);
    CH[i] = __cosf(th);
    SH[i] = __sinf(th);
  }
  for (int i = tid; i < 8192; i += 256) {
    int h = i >> 6, j = i & 63;
    float f = (j < 32) ? (float)j : (float)(j + 64);
    float th = PI2 * (float)h * f * (1.0f / 128.0f);
    CI[i] = __cosf(th) * (1.0f / 128.0f);
    SI[i] = __sinf(th) * (1.0f / 128.0f);
  }
  for (int i = tid; i < 8192; i += 256) {
    int k = i >> 7, w = i & 127;
    int kk = k & 31;
    float a = (kk == 0) ? (1.0f / 128.0f) : (2.0f / 128.0f);
    float th = PI2 * (float)w * (float)kk * (1.0f / 128.0f);
    FI[i] = (k < 32) ? a * __cosf(th) : -a * __sinf(th);
  }
}

// ---------------------------------------------------------------------------
// Kernel 1: forward rDFT along W, keep 32 modes.
// rows = B*C*H = 131072, 64 rows/WG, 4 waves x 16 rows, N=64 (re|im), K=128.
// ---------------------------------------------------------------------------
__global__ __launch_bounds__(128) void k_wdft(const float* __restrict__ x,
                                              const float* __restrict__ tab,
                                              float* __restrict__ yfr,
                                              float* __restrict__ yfi) {
  __shared__ float sF[128 * 64];  // table [k][n]
  __shared__ float sA[64 * 128];  // 64 rows x 128 cols
  int tid = threadIdx.x, lane = tid & 31, wv = tid >> 5;
  for (int i = tid; i < 8192; i += 128) sF[i] = tab[i];
  size_t base = (size_t)blockIdx.x * 64 * 128;
  for (int i = tid; i < 8192; i += 128) sA[i] = x[base + i];
  __syncthreads();

  const float* Aw = sA + wv * 16 * 128;
  v8f acc[4];
#pragma unroll
  for (int nt = 0; nt < 4; ++nt) acc[nt] = v8zero();
  for (int kt = 0; kt < 32; ++kt) {
    v2f a = ldA(Aw, 128, 0, kt * 4, lane);
#pragma unroll
    for (int nt = 0; nt < 4; ++nt) {
      v2f b = ldB(sF, 64, kt * 4, nt * 16, lane);
      acc[nt] = wmma4(a, b, acc[nt]);
    }
  }
  size_t row0 = (size_t)blockIdx.x * 64 + wv * 16;
  int half = (lane >> 4) & 1, n = lane & 15;
#pragma unroll
  for (int nt = 0; nt < 4; ++nt) {
    float* dst = (nt < 2) ? yfr : yfi;
    int col = (nt & 1) * 16 + n;
#pragma unroll
    for (int g = 0; g < 8; ++g) {
      size_t r = row0 + g + half * 8;
      dst[r * 32 + col] = acc[nt][g];
    }
  }
}

// ---------------------------------------------------------------------------
// Kernel 2: forward complex DFT along H, keep 64 rows (kh 0..31, 96..127).
// One WG per (b,c). Zr = CH*Yr + SH*Yi ; Zi = CH*Yi - SH*Yr.
// Tables read from global (L2-hot), Y staged in LDS.
// ---------------------------------------------------------------------------
__global__ __launch_bounds__(256) void k_hdft(const float* __restrict__ yfr,
                                              const float* __restrict__ yfi,
                                              const float* __restrict__ CH,
                                              const float* __restrict__ SH,
                                              float* __restrict__ zfr,
                                              float* __restrict__ zfi) {
  __shared__ float sYr[128 * 32];
  __shared__ float sYi[128 * 32];
  int tid = threadIdx.x, lane = tid & 31, wv = tid >> 5;
  size_t bc = blockIdx.x;
  const float* pr = yfr + bc * 4096;
  const float* pi = yfi + bc * 4096;
  for (int i = tid; i < 4096; i += 256) {
    sYr[i] = pr[i];
    sYi[i] = pi[i];
  }
  __syncthreads();
#pragma unroll
  for (int it = 0; it < 2; ++it) {
    int job = wv + 8 * it;
    int part = job & 1, mt = (job >> 1) & 3, nt = (job >> 3) & 1;
    v8f acc = v8zero();
    for (int kt = 0; kt < 32; ++kt) {
      int k0 = kt * 4;
      v2f aC = ldA(CH, 128, mt * 16, k0, lane);
      v2f aS = ldA(SH, 128, mt * 16, k0, lane);
      if (part == 0) {
        v2f bR = ldB(sYr, 32, k0, nt * 16, lane);
        v2f bI = ldB(sYi, 32, k0, nt * 16, lane);
        acc = wmma4(aC, bR, acc);
        acc = wmma4(aS, bI, acc);
      } else {
        v2f bI = ldB(sYi, 32, k0, nt * 16, lane);
        v2f bRn = ldB(sYr, 32, k0, nt * 16, lane);
        bRn.x = -bRn.x;
        bRn.y = -bRn.y;
        acc = wmma4(aC, bI, acc);
        acc = wmma4(aS, bRn, acc);
      }
    }
    float* dst = part ? zfi : zfr;
    stC(dst + bc * 2048, 32, mt * 16, nt * 16, lane, acc);
  }
}

// ---------------------------------------------------------------------------
// Kernel 3: spectral channel mix (per-mode complex channel contraction).
// Mode dim is innermost in both xf and weights -> coalesced VALU FMA.
// out[b,o,j,kw] = sum_c xf[b,c,j,kw] * w[corner(j), c, o, kh(j), kw]
// ---------------------------------------------------------------------------
__global__ __launch_bounds__(256) void k_mix(const float* __restrict__ zfr,
                                             const float* __restrict__ zfi,
                                             const float* __restrict__ wr,
                                             const float* __restrict__ wi,
                                             float* __restrict__ mxr,
                                             float* __restrict__ mxi) {
  int b = blockIdx.x >> 7;
  int o = blockIdx.x & 127;
  int tid = threadIdx.x;
#pragma unroll 1
  for (int it = 0; it < 8; ++it) {
    int m = it * 256 + tid;          // 0..2047
    int j = m >> 5, kw = m & 31;
    int corner = j >> 5, kh = j & 31;
    float accR = 0.f, accI = 0.f;
    int xbase = b * 262144 + j * 32 + kw;           // + c*2048
    int wbase = corner * 16777216 + o * 1024 + kh * 32 + kw;  // + c*131072
#pragma unroll 4
    for (int c = 0; c < 128; ++c) {
      float xr = zfr[xbase + c * 2048];
      float xi = zfi[xbase + c * 2048];
      float wrv = wr[wbase + c * 131072];
      float wiv = wi[wbase + c * 131072];
      accR = fmaf(xr, wrv, fmaf(-xi, wiv, accR));
      accI = fmaf(xr, wiv, fmaf(xi, wrv, accI));
    }
    int oidx = (b * 128 + o) * 2048 + m;
    mxr[oidx] = accR;
    mxi[oidx] = accI;
  }
}

// ---------------------------------------------------------------------------
// Kernel 4: inverse DFT along H (64 modes -> 128 rows), 1/H folded in tables.
// Z'r = CI*zr - SI*zi ; Z'i = SI*zr + CI*zi.  One WG per (b,o).
// ---------------------------------------------------------------------------
__global__ __launch_bounds__(256) void k_ihdft(const float* __restrict__ mxr,
                                               const float* __restrict__ mxi,
                                               const float* __restrict__ CI,
                                               const float* __restrict__ SI,
                                               float* __restrict__ zpr,
                                               float* __restrict__ zpi) {
  __shared__ float sZr[64 * 32];
  __shared__ float sZi[64 * 32];
  int tid = threadIdx.x, lane = tid & 31, wv = tid >> 5;
  size_t bo = blockIdx.x;
  const float* pr = mxr + bo * 2048;
  const float* pi = mxi + bo * 2048;
  for (int i = tid; i < 2048; i += 256) {
    sZr[i] = pr[i];
    sZi[i] = pi[i];
  }
  __syncthreads();
#pragma unroll
  for (int it = 0; it < 4; ++it) {
    int job = wv + 8 * it;                       // 0..31
    int part = job & 1, mt = (job >> 1) & 7, nt = (job >> 4) & 1;
    v8f acc = v8zero();
    for (int kt = 0; kt < 16; ++kt) {
      int k0 = kt * 4;
      v2f aC = ldA(CI, 64, mt * 16, k0, lane);
      v2f aS = ldA(SI, 64, mt * 16, k0, lane);
      v2f bR = ldB(sZr, 32, k0, nt * 16, lane);
      v2f bI = ldB(sZi, 32, k0, nt * 16, lane);
      if (part == 0) {
        v2f bIn = bI;
        bIn.x = -bIn.x;
        bIn.y = -bIn.y;
        acc = wmma4(aC, bR, acc);
        acc = wmma4(aS, bIn, acc);
      } else {
        acc = wmma4(aS, bR, acc);
        acc = wmma4(aC, bI, acc);
      }
    }
    float* dst = part ? zpi : zpr;
    stC(dst + bo * 4096, 32, mt * 16, nt * 16, lane, acc);
  }
}

// ---------------------------------------------------------------------------
// Kernel 5: inverse rDFT along W (64 re|im inputs -> 128 reals) + GELU.
// rows = B*O*H = 131072, 128 rows/WG (8 waves x 16), table in LDS.
// ---------------------------------------------------------------------------
__global__ __launch_bounds__(256) void k_irfft(const float* __restrict__ zpr,
                                               const float* __restrict__ zpi,
                                               const float* __restrict__ FI,
                                               float* __restrict__ y) {
  __shared__ float sT[64 * 128];  // [k][w]
  __shared__ float sZ[128 * 64];  // 128 rows x (re32|im32)
  int tid = threadIdx.x, lane = tid & 31, wv = tid >> 5;
  for (int i = tid; i < 8192; i += 256) sT[i] = FI[i];
  size_t row0g = (size_t)blockIdx.x * 128;
  for (int i = tid; i < 8192; i += 256) {
    int r = i >> 6, col = i & 63;
    size_t grow = row0g + r;
    sZ[i] = (col < 32) ? zpr[grow * 32 + col] : zpi[grow * 32 + (col - 32)];
  }
  __syncthreads();
  const float* Az = sZ + wv * 16 * 64;
  v8f acc[8];
#pragma unroll
  for (int nt = 0; nt < 8; ++nt) acc[nt] = v8zero();
  for (int kt = 0; kt < 16; ++kt) {
    v2f a = ldA(Az, 64, 0, kt * 4, lane);
#pragma unroll
    for (int nt = 0; nt < 8; ++nt) {
      v2f b = ldB(sT, 128, kt * 4, nt * 16, lane);
      acc[nt] = wmma4(a, b, acc[nt]);
    }
  }
  size_t row0 = row0g + wv * 16;
  int half = (lane >> 4) & 1, n = lane & 15;
#pragma unroll
  for (int nt = 0; nt < 8; ++nt) {
    int col = nt * 16 + n;
#pragma unroll
    for (int g = 0; g < 8; ++g) {
      size_t r = row0 + g + half * 8;
      y[r * 128 + col] = gelu_exact(acc[nt][g]);
    }
  }
}

// ---------------------------------------------------------------------------
// Kernel 6: fused channel MLP: out = gelu(W2 * gelu(W1 * Y + b1) + b2).
// 64-pixel tiles; weights (128 KB total) shared by all WGs -> L2 resident.
// ---------------------------------------------------------------------------
__global__ __launch_bounds__(256) void k_mlp(const float* __restrict__ y,
                                             const float* __restrict__ w1,
                                             const float* __restrict__ b1,
                                             const float* __restrict__ w2,
                                             const float* __restrict__ b2,
                                             float* __restrict__ out) {
  __shared__ float sY[128 * 64];
  __shared__ float sH[128 * 64];
  int tid = threadIdx.x, lane = tid & 31, wv = tid >> 5;
  size_t P0 = (size_t)blockIdx.x * 64;
  int b = (int)(P0 >> 14);        // 16384 pixels per batch image
  int s0 = (int)(P0 & 16383);
  const float* yb = y + (size_t)b * 128 * 16384;
  for (int i = tid; i < 8192; i += 256) {
    int c = i >> 6, pl = i & 63;
    sY[i] = yb[(size_t)c * 16384 + s0 + pl];
  }
  __syncthreads();
  // layer 1: H = gelu(W1 * Y + b1)
#pragma unroll
  for (int it = 0; it < 4; ++it) {
    int job = wv + 8 * it;                  // 0..31
    int mt = job & 7, nt = (job >> 3) & 3;
    v8f acc = v8zero();
    for (int kt = 0; kt < 32; ++kt) {
      v2f a = ldA(w1, 128, mt * 16, kt * 4, lane);
      v2f bb = ldB(sY, 64, kt * 4, nt * 16, lane);
      acc = wmma4(a, bb, acc);
    }
    int half = (lane >> 4) & 1, n = lane & 15;
#pragma unroll
    for (int g = 0; g < 8; ++g) {
      int row = mt * 16 + g + half * 8;
      sH[row * 64 + nt * 16 + n] = gelu_exact(acc[g] + b1[row]);
    }
  }
  __syncthreads();
  // layer 2: OUT = gelu(W2 * H + b2)
  float* ob = out + (size_t)b * 128 * 16384;
#pragma unroll
  for (int it = 0; it < 4; ++it) {
    int job = wv + 8 * it;
    int mt = job & 7, nt = (job >> 3) & 3;
    v8f acc = v8zero();
    for (int kt = 0; kt < 32; ++kt) {
      v2f a = ldA(w2, 128, mt * 16, kt * 4, lane);
      v2f bb = ldB(sH, 64, kt * 4, nt * 16, lane);
      acc = wmma4(a, bb, acc);
    }
    int half = (lane >> 4) & 1, n = lane & 15;
#pragma unroll
    for (int g = 0; g < 8; ++g) {
      int row = mt * 16 + g + half * 8;
      ob[(size_t)row * 16384 + s0 + nt * 16 + n] = gelu_exact(acc[g] + b2[row]);
    }
  }
}

// ---------------------------------------------------------------------------
// Host-side launcher
// ---------------------------------------------------------------------------
extern "C" void kernel_launch(void* const* d_in, const int* in_sizes, int n_in,
                              void* d_out, int out_size, void* d_ws,
                              size_t ws_size, hipStream_t stream) {
  (void)in_sizes;
  (void)n_in;
  (void)out_size;
  (void)ws_size;
  const float* x = (const float*)d_in[0];
  const float* wr = (const float*)d_in[1];
  const float* wi = (const float*)d_in[2];
  const float* w1 = (const float*)d_in[3];
  const float* b1 = (const float*)d_in[4];
  const float* w2 = (const float*)d_in[5];
  const float* b2 = (const float*)d_in[6];
  float* out = (float*)d_out;
  float* ws = (float*)d_ws;

  float* Ffwd = ws + OFF_TAB;
  float* CH = ws + 8192;
  float* SH = ws + 16384;
  float* CI = ws + 24576;
  float* SI = ws + 32768;
  float* FI = ws + 40960;
  float* yfr = ws + OFF_YFR;
  float* yfi = ws + OFF_YFI;
  float* zfr = ws + OFF_ZFR;
  float* zfi = ws + OFF_ZFI;
  float* mxr = ws + OFF_MXR;
  float* mxi = ws + OFF_MXI;
  float* zpr = yfr;  // reuse: yf dead after k_hdft
  float* zpi = yfi;
  float* ybuf = ws + OFF_Y;

  k_tables<<<1, 256, 0, stream>>>(ws);
  k_wdft<<<2048, 128, 0, stream>>>(x, Ffwd, yfr, yfi);
  k_hdft<<<1024, 256, 0, stream>>>(yfr, yfi, CH, SH, zfr, zfi);
  k_mix<<<1024, 256, 0, stream>>>(zfr, zfi, wr, wi, mxr, mxi);
  k_ihdft<<<1024, 256, 0, stream>>>(mxr, mxi, CI, SI, zpr, zpi);
  k_irfft<<<1024, 256, 0, stream>>>(zpr, zpi, FI, ybuf);
  k_mlp<<<2048, 256, 0, stream>>>(ybuf, w1, b1, w2, b2, out);
}